// ProximityAwareLoss_3315714753089
// MI455X (gfx1250) — compile-verified
//
#include <hip/hip_runtime.h>

// ProximityAwareLoss for N=8192, D=2, f32.
//
// d2_ij = ||pi||^2 + ||pj||^2 - 2 pi.pj is computed 16x16-tile-at-a-time with
// a single V_WMMA_F32_16X16X4_F32 by augmenting K=4:
//   A row i = [-2x_i, -2y_i, sq_i, 1],  B col j = [x_j, y_j, 1, sq_j]
// so D = A x B yields d2 directly (no epilogue adds needed).
//
// Tile loop is scalarized with readfirstlane (wave-uniform by construction) so
// the triangle tests are SALU s_cbranch, not exec-mask ops. Epilogue per
// element: raw v_max_num (inline asm, skips NaN-canonicalize), raw v_sqrt_f32,
// sub, max0, fmac -> 4 VALU + 1 trans per pair.

#define N_PTS   8192
#define MIN_D   0.1f
#define TILES1D (N_PTS / 16)              // 512
#define TILESSQ (TILES1D * TILES1D)       // 262144
#define NBLK1   1024
#define THREADS 256
#define WPB     (THREADS / 32)            // 8 waves per block (wave32)
#define NWAVES  (NBLK1 * WPB)             // 8192 waves -> 32 tiles/wave

typedef float v2f __attribute__((ext_vector_type(2)));
typedef float v8f __attribute__((ext_vector_type(8)));

// Single v_max_num_f32 without the NaN-canonicalize the fmaxf lowering adds.
// (WMMA output of finite inputs is never NaN, so canonicalization is dead.)
__device__ __forceinline__ float clamp0(float x) {
    float r;
    asm("v_max_num_f32 %0, 0, %1" : "=v"(r) : "v"(x));
    return r;
}

__global__ void __launch_bounds__(THREADS)
prox_pairs_kernel(const float* __restrict__ pred, float* __restrict__ blockSums) {
    const int lane = threadIdx.x & 31;
    const int wave = threadIdx.x >> 5;
    const int l15  = lane & 15;
    const bool hi  = lane >= 16;

    // Wave-uniform tile index, forced into SGPRs so the tile loop is scalar.
    const int tile0 = __builtin_amdgcn_readfirstlane(blockIdx.x * WPB + wave);

    float acc = 0.0f;

    for (int tile = tile0; tile < TILESSQ; tile += NWAVES) {
        const int tj = tile >> 9;              // tile / 512   (scalar)
        const int ti = tile & (TILES1D - 1);   // tile % 512   (scalar)
        if (ti > tj) continue;                 // s_cbranch: EXEC untouched

        const int i0 = ti * 16;
        const int j0 = tj * 16;

        // Points are fully cache-resident (64KB total) -> cheap reloads.
        const float2 pi = ((const float2*)pred)[i0 + l15];
        const float2 pj = ((const float2*)pred)[j0 + l15];
        const float sqi = pi.x * pi.x + pi.y * pi.y;
        const float sqj = pj.x * pj.x + pj.y * pj.y;

        // A (16x4 f32): lanes 0-15 hold K0/K1, lanes 16-31 hold K2/K3 (per ISA layout)
        v2f a, b;
        a.x = hi ? sqi  : -2.0f * pi.x;   // K0 = -2x | K2 = sq
        a.y = hi ? 1.0f : -2.0f * pi.y;   // K1 = -2y | K3 = 1
        // B (4x16 f32): VGPR0 = rows K0|K2 across N, VGPR1 = rows K1|K3
        b.x = hi ? 1.0f : pj.x;           // K0 = x  | K2 = 1
        b.y = hi ? sqj  : pj.y;           // K1 = y  | K3 = sq

        v8f c = {};
        // (neg_a, A, neg_b, B, c_mod, C, reuse_a, reuse_b)
        c = __builtin_amdgcn_wmma_f32_16x16x4_f32(
                false, a, false, b, (short)0, c, false, false);

        // C/D layout: VGPR r -> M = r (lanes 0-15) / r+8 (lanes 16-31); N = lane&15
        if (ti != tj) {
            // Off-diagonal tile: i < j guaranteed for all 256 pairs -> no mask.
#pragma unroll
            for (int r = 0; r < 8; ++r) {
                const float d2 = clamp0(c[r]);
                const float t  = fmaxf(MIN_D - __builtin_amdgcn_sqrtf(d2), 0.0f);
                acc = fmaf(t, t, acc);
            }
        } else {
            // Diagonal tile: mask to strict upper triangle within the tile.
            const int j     = l15;
            const int ibase = hi ? 8 : 0;
#pragma unroll
            for (int r = 0; r < 8; ++r) {
                const float d2 = clamp0(c[r]);
                const float t  = fmaxf(MIN_D - __builtin_amdgcn_sqrtf(d2), 0.0f);
                acc += ((ibase + r) < j) ? t * t : 0.0f;
            }
        }
    }

    // wave32 butterfly reduction
#pragma unroll
    for (int off = 16; off > 0; off >>= 1)
        acc += __shfl_xor(acc, off, 32);

    __shared__ float wsum[WPB];
    if (lane == 0) wsum[wave] = acc;
    __syncthreads();
    if (threadIdx.x == 0) {
        float s = 0.0f;
#pragma unroll
        for (int w = 0; w < WPB; ++w) s += wsum[w];
        blockSums[blockIdx.x] = s;
    }
}

// Single-block deterministic finalize: sum 1024 partials + MSE term.
__global__ void __launch_bounds__(THREADS)
prox_finalize_kernel(const float* __restrict__ pred, const float* __restrict__ targ,
                     const float* __restrict__ blockSums, float* __restrict__ out) {
    __shared__ float sPen[THREADS];
    __shared__ float sMse[THREADS];
    const int tid = threadIdx.x;

    float p = 0.0f;
    for (int idx = tid; idx < NBLK1; idx += THREADS) p += blockSums[idx];

    float m = 0.0f;
    for (int idx = tid; idx < N_PTS * 2; idx += THREADS) {
        const float d = pred[idx] - targ[idx];
        m += d * d;
    }

    sPen[tid] = p;
    sMse[tid] = m;
    __syncthreads();
    for (int off = THREADS / 2; off > 0; off >>= 1) {
        if (tid < off) {
            sPen[tid] += sPen[tid + off];
            sMse[tid] += sMse[tid + off];
        }
        __syncthreads();
    }
    if (tid == 0)
        out[0] = sMse[0] * (1.0f / (float)(N_PTS * 2)) + sPen[0];
}

extern "C" void kernel_launch(void* const* d_in, const int* in_sizes, int n_in,
                              void* d_out, int out_size, void* d_ws, size_t ws_size,
                              hipStream_t stream) {
    (void)in_sizes; (void)n_in; (void)out_size; (void)ws_size;
    const float* pred = (const float*)d_in[0];
    const float* targ = (const float*)d_in[1];
    float* blockSums  = (float*)d_ws;          // NBLK1 floats = 4KB
    float* out        = (float*)d_out;

    prox_pairs_kernel<<<NBLK1, THREADS, 0, stream>>>(pred, blockSums);
    prox_finalize_kernel<<<1, THREADS, 0, stream>>>(pred, targ, blockSums, out);
}